// get_loss_87222195848289
// MI455X (gfx1250) — compile-verified
//
#include <hip/hip_runtime.h>

typedef __attribute__((ext_vector_type(2))) float v2f;
typedef __attribute__((ext_vector_type(8))) float v8f;

#define NPTS 4096
#define NB   8

// Zero the [B,2] chamfer accumulator in workspace (harness poisons ws).
__global__ void zero_ws_kernel(float* __restrict__ ws) {
    if (threadIdx.x < NB * 2) ws[threadIdx.x] = 0.0f;
}

// One chamfer direction: for each point in P1, min squared distance to any
// point in P2; mean over P1's points accumulated into chamfer[b*2 + dir].
// P1,P2: [B,3,NPTS].  grid = (NPTS/128, B), block = 256 (8 waves of 32).
// Each wave owns a 16-row A tile and sweeps all 256 column tiles of P2.
// WMMA folding: A row k=0..3 = (-2x0,-2x1,-2x2,|x|^2), B col = (y0,y1,y2,1)
//   => C = |x|^2 - 2 x.y ; the per-column |y|^2 is added in the min step.
__global__ __launch_bounds__(256) void chamfer_dir_kernel(
    const float* __restrict__ P1, const float* __restrict__ P2,
    float* __restrict__ chamfer, int dir)
{
    const int b    = blockIdx.y;
    const int wave = threadIdx.x >> 5;
    const int lane = threadIdx.x & 31;
    const int half = lane >> 4;   // 0: lanes 0-15 (K=0,1), 1: lanes 16-31 (K=2,3)
    const int l16  = lane & 15;

    const float* p1 = P1 + (size_t)b * 3 * NPTS;
    const float* p2 = P2 + (size_t)b * 3 * NPTS;

    // ---- A tile: 16 rows of p1 (ISA 16x4 f32 A layout) ----
    const int row = blockIdx.x * 128 + wave * 16 + l16;
    const float x0 = p1[0 * NPTS + row];
    const float x1 = p1[1 * NPTS + row];
    const float x2 = p1[2 * NPTS + row];
    const float nx = x0 * x0 + x1 * x1 + x2 * x2;
    v2f a;
    if (half == 0) { a.x = -2.0f * x0; a.y = -2.0f * x1; }
    else           { a.x = -2.0f * x2; a.y = nx;         }

    // running min per C element: rowmin[v] covers row (v + 8*half), col l16
    v8f rowmin;
#pragma unroll
    for (int v = 0; v < 8; ++v) rowmin[v] = 3.0e38f;

    for (int mt = 0; mt < NPTS / 16; ++mt) {
        const int col = mt * 16 + l16;
        const float y0 = p2[0 * NPTS + col];
        const float y1 = p2[1 * NPTS + col];
        const float y2 = p2[2 * NPTS + col];
        const float ny = y0 * y0 + y1 * y1 + y2 * y2;   // per-lane column norm
        v2f bv;
        if (half == 0) { bv.x = y0; bv.y = y1;   }
        else           { bv.x = y2; bv.y = 1.0f; }

        v8f c = {};
        // (neg_a, A, neg_b, B, c_mod, C, reuse_a, reuse_b)
        c = __builtin_amdgcn_wmma_f32_16x16x4_f32(
            false, a, false, bv, (short)0, c, false, false);

#pragma unroll
        for (int v = 0; v < 8; ++v)
            rowmin[v] = fminf(rowmin[v], c[v] + ny);    // + |y_col|^2
    }

    // min-reduce across the 16 lanes (columns) inside each half (butterfly)
#pragma unroll
    for (int mask = 1; mask < 16; mask <<= 1) {
#pragma unroll
        for (int v = 0; v < 8; ++v)
            rowmin[v] = fminf(rowmin[v], __shfl_xor(rowmin[v], mask, 32));
    }
    // every lane of a half now holds the row-min for rows v + 8*half
    float s = 0.0f;
#pragma unroll
    for (int v = 0; v < 8; ++v) s += rowmin[v];   // sum 8 rows of this half
    s += __shfl_xor(s, 16, 32);                   // + other half's 8 rows

    if (lane == 0)
        atomicAdd(&chamfer[b * 2 + dir], s * (1.0f / NPTS));
}

// Final scalar: mean((pred - chamfer)^2) over all 16 elements.
__global__ void mse_kernel(const float* __restrict__ pred,
                           const float* __restrict__ chamfer,
                           float* __restrict__ out)
{
    float acc = 0.0f;
    for (int i = 0; i < NB * 2; ++i) {
        const float d = pred[i] - chamfer[i];
        acc += d * d;
    }
    out[0] = acc * (1.0f / (NB * 2));
}

extern "C" void kernel_launch(void* const* d_in, const int* in_sizes, int n_in,
                              void* d_out, int out_size, void* d_ws, size_t ws_size,
                              hipStream_t stream) {
    (void)in_sizes; (void)n_in; (void)out_size; (void)ws_size;
    const float* pred         = (const float*)d_in[0];   // [B,2]
    const float* points       = (const float*)d_in[1];   // [B,3,N]
    const float* points_trans = (const float*)d_in[2];   // [B,3,N]
    float* chamfer = (float*)d_ws;                       // [B,2] accumulator
    float* out     = (float*)d_out;                      // scalar

    zero_ws_kernel<<<1, 32, 0, stream>>>(chamfer);

    dim3 grid(NPTS / 128, NB);
    // dir 0: points -> points_trans (d1 means)
    chamfer_dir_kernel<<<grid, 256, 0, stream>>>(points, points_trans, chamfer, 0);
    // dir 1: points_trans -> points (d2 means)
    chamfer_dir_kernel<<<grid, 256, 0, stream>>>(points_trans, points, chamfer, 1);

    mse_kernel<<<1, 1, 0, stream>>>(pred, chamfer, out);
}